// MultiFrequencyFourierKernel_46566035423348
// MI455X (gfx1250) — compile-verified
//
#include <hip/hip_runtime.h>
#include <hip/hip_bf16.h>
#include <math.h>

typedef __attribute__((ext_vector_type(2))) float v2f;
typedef __attribute__((ext_vector_type(8))) float v8f;

#define B_  8
#define N_  512
#define M_  512
#define P_  3
#define NF  8
#define NTI (N_ / 16)   // 32 n-tiles
#define MTI (M_ / 16)   // 32 m-tiles

#define INV_2PI 0.15915494309189535f   // 1/(2*pi): HW v_sin/v_cos take revolutions

// One wave32 computes one 16x16 (n,m) tile of distances via
// V_WMMA_F32_16X16X4_F32 (K=3 padded to 4), then a raw-HW sincos epilogue.
__global__ __launch_bounds__(256) void MultiFrequencyFourierKernel_46566035423348_kernel(
    const float* __restrict__ z,      // [M, 3]
    const float* __restrict__ mu,     // [B, N, 3]
    const float* __restrict__ freqs,  // [8]
    float* __restrict__ out)          // [B, N, M, 16]
{
    const int wave = threadIdx.x >> 5;
    const int lane = threadIdx.x & 31;
    const int l    = lane & 15;
    const int half = lane >> 4;       // 0: K={0,1}, 1: K={2,3(pad)}

    int tile = blockIdx.x * 8 + wave;           // 0 .. 8191
    const int mi = tile & (MTI - 1); tile >>= 5;
    const int ni = tile & (NTI - 1); tile >>= 5;
    const int b  = tile;

    const int n0 = ni * 16;
    const int m0 = mi * 16;

    // Clipped frequencies, pre-scaled to revolutions (uniform -> scalar loads)
    float fr[NF];
#pragma unroll
    for (int f = 0; f < NF; ++f)
        fr[f] = fminf(fmaxf(freqs[f], 0.1f), 10.0f) * INV_2PI;

    // ---- A matrix: mu tile, 16x4 (K padded with zero) ----
    // ISA layout: lanes 0-15 -> {K0,K1}, lanes 16-31 -> {K2,K3}, row M = lane&15
    const float* murow = mu + (size_t)(b * N_ + n0 + l) * P_;
    const float mu0 = murow[0], mu1 = murow[1], mu2 = murow[2];
    v2f a;
    a.x = half ? mu2 : mu0;
    a.y = half ? 0.0f : mu1;

    // ---- B matrix: z tile, 4x16 (mirrored layout), col N = lane&15 ----
    const float* zrow = z + (size_t)(m0 + l) * P_;
    const float z0 = zrow[0], z1 = zrow[1], z2 = zrow[2];
    v2f bv;
    bv.x = half ? z2 : z0;
    bv.y = half ? 0.0f : z1;

    // Gram tile: dot[n][m] = sum_k mu[n,k] * z[m,k]
    v8f acc = {};
    acc = __builtin_amdgcn_wmma_f32_16x16x4_f32(
        /*neg_a=*/false, a, /*neg_b=*/false, bv,
        /*c_mod=*/(short)0, acc, /*reuse_a=*/false, /*reuse_b=*/false);

    // Rank-1 norm corrections
    const float nmu = mu0 * mu0 + mu1 * mu1 + mu2 * mu2;  // ||mu[n0+l]||^2
    const float nz  = z0 * z0 + z1 * z1 + z2 * z2;        // ||z[m0+l]||^2
    const int mcol  = m0 + l;

#pragma unroll
    for (int g = 0; g < 8; ++g) {
        const int row  = g + 8 * half;                 // local n row 0..15
        const float nr = __shfl(nmu, row, 32);         // ||mu[n0+row]||^2
        const float d2 = nr + nz - 2.0f * acc[g];
        // raw v_sqrt_f32: args are well-scaled, skip the libm fixup sequence
        const float dist = __builtin_amdgcn_sqrtf(fmaxf(d2, 0.0f));

        float o[2 * NF];
#pragma unroll
        for (int f = 0; f < NF; ++f) {
            const float ph = dist * fr[f];             // revolutions
            o[f]      = __builtin_amdgcn_cosf(ph);     // raw v_cos_f32
            o[f + NF] = __builtin_amdgcn_sinf(ph);     // raw v_sin_f32
        }

        const size_t base =
            (((size_t)b * N_ + (n0 + row)) * M_ + mcol) * (size_t)(2 * NF);
        float4* op = (float4*)(out + base);
        op[0] = make_float4(o[0],  o[1],  o[2],  o[3]);
        op[1] = make_float4(o[4],  o[5],  o[6],  o[7]);
        op[2] = make_float4(o[8],  o[9],  o[10], o[11]);
        op[3] = make_float4(o[12], o[13], o[14], o[15]);
    }
}

extern "C" void kernel_launch(void* const* d_in, const int* in_sizes, int n_in,
                              void* d_out, int out_size, void* d_ws, size_t ws_size,
                              hipStream_t stream) {
    const float* z     = (const float*)d_in[0];  // [512,3]
    const float* mu    = (const float*)d_in[1];  // [8,512,3]
    // d_in[2] = sigma, unused (interface compat with reference)
    const float* freqs = (const float*)d_in[3];  // [8]
    float* out = (float*)d_out;                  // [8,512,512,16]

    const int tiles  = B_ * NTI * MTI;           // 8192 waves
    const int blocks = tiles / 8;                // 8 waves per 256-thread block
    MultiFrequencyFourierKernel_46566035423348_kernel<<<blocks, 256, 0, stream>>>(
        z, mu, freqs, out);
}